// AMSLossv1_996432413429
// MI455X (gfx1250) — compile-verified
//
#include <hip/hip_runtime.h>
#include <hip/hip_bf16.h>

typedef __attribute__((ext_vector_type(2))) float v2f;
typedef __attribute__((ext_vector_type(8))) float v8f;

#define NROWS 16384
#define NCOLS 10000
#define AMS_M 0.4f
#define AMS_EPS 1e-10f

// Full wave32 sum via V_WMMA_F32_16X16X4_F32.
// A-matrix gets the 32 per-lane partials (a0 = s, a1 = 0), B = all ones.
// D[m][n] = rowsum_A(m); summing one full column of D (all 16 M values for a
// fixed n) == sum of all A elements == sum over all 32 lanes. A lane holds
// 8 D values covering 8 of the 16 rows; its lane^16 partner holds the other 8.
__device__ __forceinline__ float wave_sum(float s) {
#if __has_builtin(__builtin_amdgcn_wmma_f32_16x16x4_f32)
    v2f a; a[0] = s;    a[1] = 0.0f;
    v2f b; b[0] = 1.0f; b[1] = 1.0f;
    v8f c = {};
    c = __builtin_amdgcn_wmma_f32_16x16x4_f32(
        /*neg_a=*/false, a, /*neg_b=*/false, b,
        /*c_mod=*/(short)0, c, /*reuse_a=*/false, /*reuse_b=*/false);
    float t = ((c[0] + c[1]) + (c[2] + c[3])) + ((c[4] + c[5]) + (c[6] + c[7]));
    return t + __shfl_xor(t, 16, 32);
#else
    #pragma unroll
    for (int off = 16; off > 0; off >>= 1) s += __shfl_xor(s, off, 32);
    return s;
#endif
}

// One workgroup (8 waves) per row: stream 10000 fp32 via b128 loads,
// accumulate exp() per lane, WMMA wave-reduce, LDS combine, write row loss.
__global__ void __launch_bounds__(256) ams_row_kernel(
    const float* __restrict__ logits,
    const int*   __restrict__ targets,
    float*       __restrict__ row_loss)
{
    const int row = blockIdx.x;
    const int tid = threadIdx.x;
    const float* rowp = logits + (size_t)row * NCOLS;
    const float4* rp4 = (const float4*)rowp;   // 40000 B row stride -> 16B aligned

    // 2500 float4 per row = 9 full sweeps of 256 threads + tail of 196.
    float s = 0.0f;
    int i = tid;
    #pragma unroll
    for (int it = 0; it < 9; ++it) {
        float4 v = rp4[i];
        s += (__expf(v.x) + __expf(v.y)) + (__expf(v.z) + __expf(v.w));
        i += 256;
    }
    if (i < (NCOLS / 4)) {
        float4 v = rp4[i];
        s += (__expf(v.x) + __expf(v.y)) + (__expf(v.z) + __expf(v.w));
    }

    // All 256 threads convergent here: EXEC all-ones for WMMA.
    float ws = wave_sum(s);

    __shared__ float partial[8];
    if ((tid & 31) == 0) partial[tid >> 5] = ws;
    __syncthreads();

    if (tid == 0) {
        float S = 0.0f;
        #pragma unroll
        for (int w = 0; w < 8; ++w) S += partial[w];
        const float num  = rowp[targets[row]];
        const float en   = __expf(num);
        const float expm = __expf(AMS_M);
        const float denom = en + (S * expm - en * expm);
        row_loss[row] = num - __logf(denom + AMS_EPS);
    }
}

// Deterministic fixed-tree reduction of 16384 row losses -> scalar -mean.
__global__ void __launch_bounds__(256) ams_final_kernel(
    const float* __restrict__ row_loss,
    float*       __restrict__ out)
{
    const int tid = threadIdx.x;
    float s = 0.0f;
    #pragma unroll 4
    for (int it = 0; it < NROWS / 256; ++it)   // 64 coalesced sweeps
        s += row_loss[it * 256 + tid];

    float ws = wave_sum(s);

    __shared__ float partial[8];
    if ((tid & 31) == 0) partial[tid >> 5] = ws;
    __syncthreads();

    if (tid == 0) {
        float total = 0.0f;
        #pragma unroll
        for (int w = 0; w < 8; ++w) total += partial[w];
        out[0] = -total / (float)NROWS;
    }
}

extern "C" void kernel_launch(void* const* d_in, const int* in_sizes, int n_in,
                              void* d_out, int out_size, void* d_ws, size_t ws_size,
                              hipStream_t stream)
{
    const float* logits  = (const float*)d_in[0];
    const int*   targets = (const int*)d_in[1];
    float* row_loss = (float*)d_ws;     // 16384 * 4 B = 64 KB scratch
    float* out      = (float*)d_out;

    ams_row_kernel<<<NROWS, 256, 0, stream>>>(logits, targets, row_loss);
    ams_final_kernel<<<1, 256, 0, stream>>>(row_loss, out);
}